// NodeBlock_1855425872042
// MI455X (gfx1250) — compile-verified
//
#include <hip/hip_runtime.h>

#define N_NODES_C    100000
#define N_EDGES_C    1600000
#define NODE_DIM_C   128
#define EDGE_DIM_C   48
#define GLOBAL_DIM_C 64
#define IN_DIM_C     240   // 128 + 48 + 64
#define HIDDEN_C     256
#define OUT_DIM_C    128

typedef __attribute__((ext_vector_type(2))) float v2f;
typedef __attribute__((ext_vector_type(8))) float v8f;

// ---------------- kernel 1: zero the agg scratch ----------------
__global__ void zero_agg_kernel(float* __restrict__ agg) {
  int i = blockIdx.x * blockDim.x + threadIdx.x;
  if (i < N_NODES_C * EDGE_DIM_C) agg[i] = 0.0f;
}

// ---------------- kernel 2: scatter-add edge features ----------------
// one thread per (edge, feature); edge reads fully coalesced, atomics hit
// a 19.2MB L2-resident agg region.
__global__ void scatter_edges_kernel(const float* __restrict__ edges,
                                     const int* __restrict__ recv,
                                     float* __restrict__ agg) {
  int i = blockIdx.x * blockDim.x + threadIdx.x;
  if (i >= N_EDGES_C * EDGE_DIM_C) return;
  int e = i / EDGE_DIM_C;
  int f = i - e * EDGE_DIM_C;
  atomicAdd(&agg[recv[e] * EDGE_DIM_C + f], edges[i]);
}

// ---------------- kernel 3: fused concat + MLP via fp32 WMMA ----------------
// block = 256 threads = 8 waves; block tile = 64 nodes.
// Each wave register-blocks a 2x2 grid of 16x16 tiles: one K-loop issues
// 4 wmma per 4 ds_load_2addr (A0,A1,B0,B1), i.e. 1:1 wmma:DS-load.
// LDS: X (64x241) 61.7KB + H (64x257) 65.8KB + W (256x129) 132.1KB = 259.6KB
// (CDNA5 allows up to 320KB per workgroup).
#define XS_ (IN_DIM_C + 1)   // 241
#define HS_ (HIDDEN_C + 1)   // 257
#define WS_ 129              // W chunk: up to 256 rows x 128 cols

__global__ __launch_bounds__(256)
void mlp_wmma_kernel(const float* __restrict__ nodes,
                     const float* __restrict__ agg,
                     const int* __restrict__ batch,
                     const float* __restrict__ gg,
                     const float* __restrict__ W1,
                     const float* __restrict__ b1,
                     const float* __restrict__ W2,
                     const float* __restrict__ b2,
                     float* __restrict__ out) {
  __shared__ float X_lds[64 * XS_];
  __shared__ float H_lds[64 * HS_];
  __shared__ float W_lds[HIDDEN_C * WS_];

  const int tid    = threadIdx.x;
  const int wave   = tid >> 5;
  const int lane   = tid & 31;
  const int lhalf  = lane >> 4;   // 0 for lanes 0-15, 1 for lanes 16-31
  const int l16    = lane & 15;
  const int m_base = blockIdx.x * 64;

  // wave -> 2x2 tile block: M rows {mt0,mt0+1}*16, N cols {nt0,nt0+1}*16
  const int mt0 = (wave >> 2) * 2;    // 0 or 2
  const int nt0 = (wave & 3) * 2;     // 0,2,4,6

  // ---- stage X = [nodes | agg | globals[batch]] : 64 x 240, zero-pad ragged rows
  for (int idx = tid; idx < 64 * IN_DIM_C; idx += 256) {
    int row = idx / IN_DIM_C;
    int col = idx - row * IN_DIM_C;
    int node = m_base + row;
    float v = 0.0f;
    if (node < N_NODES_C) {
      if (col < NODE_DIM_C) {
        v = nodes[node * NODE_DIM_C + col];
      } else if (col < NODE_DIM_C + EDGE_DIM_C) {
        v = agg[node * EDGE_DIM_C + (col - NODE_DIM_C)];
      } else {
        v = gg[batch[node] * GLOBAL_DIM_C + (col - NODE_DIM_C - GLOBAL_DIM_C + GLOBAL_DIM_C - EDGE_DIM_C + EDGE_DIM_C - EDGE_DIM_C)];
      }
    }
    X_lds[row * XS_ + col] = v;
  }
  __syncthreads();

  // ---- GEMM1: H = relu(X @ W1 + b1), 2 chunks of 128 columns
  for (int nc = 0; nc < 2; ++nc) {
    for (int idx = tid; idx < IN_DIM_C * 128; idx += 256) {
      int r = idx >> 7;
      int c = idx & 127;
      W_lds[r * WS_ + c] = W1[r * HIDDEN_C + nc * 128 + c];
    }
    __syncthreads();

    v8f acc00 = {}, acc01 = {}, acc10 = {}, acc11 = {};
    const float* xrow0 = &X_lds[(mt0 * 16 + l16) * XS_];
    const float* xrow1 = xrow0 + 16 * XS_;
    for (int k = 0; k < IN_DIM_C; k += 4) {
      int ka = k + 2 * lhalf;     // A layout: VGPR0={K,K+2}, VGPR1={K+1,K+3}
      v2f a0, a1, b0, b1v;
      a0.x = xrow0[ka];      a0.y = xrow0[ka + 1];
      a1.x = xrow1[ka];      a1.y = xrow1[ka + 1];
      const float* wrow0 = &W_lds[ka * WS_];
      const float* wrow1 = &W_lds[(ka + 1) * WS_];
      b0.x  = wrow0[nt0 * 16 + l16];       b0.y  = wrow1[nt0 * 16 + l16];
      b1v.x = wrow0[nt0 * 16 + 16 + l16];  b1v.y = wrow1[nt0 * 16 + 16 + l16];
      acc00 = __builtin_amdgcn_wmma_f32_16x16x4_f32(false, a0, false, b0,  (short)0, acc00, false, false);
      acc01 = __builtin_amdgcn_wmma_f32_16x16x4_f32(false, a0, false, b1v, (short)0, acc01, false, false);
      acc10 = __builtin_amdgcn_wmma_f32_16x16x4_f32(false, a1, false, b0,  (short)0, acc10, false, false);
      acc11 = __builtin_amdgcn_wmma_f32_16x16x4_f32(false, a1, false, b1v, (short)0, acc11, false, false);
    }

    // epilogue: bias + relu -> H_lds
    {
      int col0 = nc * 128 + nt0 * 16 + l16;
      int col1 = col0 + 16;
      float bias0 = b1[col0];
      float bias1 = b1[col1];
#pragma unroll
      for (int v = 0; v < 8; ++v) {   // C/D: row = v + 8*lhalf, col = l16
        int m0 = mt0 * 16 + v + 8 * lhalf;
        int m1 = m0 + 16;
        float h;
        h = acc00[v] + bias0; H_lds[m0 * HS_ + col0] = h > 0.0f ? h : 0.0f;
        h = acc01[v] + bias1; H_lds[m0 * HS_ + col1] = h > 0.0f ? h : 0.0f;
        h = acc10[v] + bias0; H_lds[m1 * HS_ + col0] = h > 0.0f ? h : 0.0f;
        h = acc11[v] + bias1; H_lds[m1 * HS_ + col1] = h > 0.0f ? h : 0.0f;
      }
    }
    __syncthreads();
  }

  // ---- GEMM2: out = H @ W2 + b2, single pass (all 128 output columns)
  for (int idx = tid; idx < HIDDEN_C * 128; idx += 256) {
    int r = idx >> 7;
    int c = idx & 127;
    W_lds[r * WS_ + c] = W2[r * OUT_DIM_C + c];
  }
  __syncthreads();

  {
    v8f acc00 = {}, acc01 = {}, acc10 = {}, acc11 = {};
    const float* hrow0 = &H_lds[(mt0 * 16 + l16) * HS_];
    const float* hrow1 = hrow0 + 16 * HS_;
    for (int k = 0; k < HIDDEN_C; k += 4) {
      int ka = k + 2 * lhalf;
      v2f a0, a1, b0, b1v;
      a0.x = hrow0[ka];      a0.y = hrow0[ka + 1];
      a1.x = hrow1[ka];      a1.y = hrow1[ka + 1];
      const float* wrow0 = &W_lds[ka * WS_];
      const float* wrow1 = &W_lds[(ka + 1) * WS_];
      b0.x  = wrow0[nt0 * 16 + l16];       b0.y  = wrow1[nt0 * 16 + l16];
      b1v.x = wrow0[nt0 * 16 + 16 + l16];  b1v.y = wrow1[nt0 * 16 + 16 + l16];
      acc00 = __builtin_amdgcn_wmma_f32_16x16x4_f32(false, a0, false, b0,  (short)0, acc00, false, false);
      acc01 = __builtin_amdgcn_wmma_f32_16x16x4_f32(false, a0, false, b1v, (short)0, acc01, false, false);
      acc10 = __builtin_amdgcn_wmma_f32_16x16x4_f32(false, a1, false, b0,  (short)0, acc10, false, false);
      acc11 = __builtin_amdgcn_wmma_f32_16x16x4_f32(false, a1, false, b1v, (short)0, acc11, false, false);
    }

    // epilogue: bias + guarded global store
    int col0 = nt0 * 16 + l16;
    int col1 = col0 + 16;
    float bias0 = b2[col0];
    float bias1 = b2[col1];
#pragma unroll
    for (int v = 0; v < 8; ++v) {
      int m0 = mt0 * 16 + v + 8 * lhalf;
      int m1 = m0 + 16;
      int node0 = m_base + m0;
      int node1 = m_base + m1;
      if (node0 < N_NODES_C) {
        out[node0 * OUT_DIM_C + col0] = acc00[v] + bias0;
        out[node0 * OUT_DIM_C + col1] = acc01[v] + bias1;
      }
      if (node1 < N_NODES_C) {
        out[node1 * OUT_DIM_C + col0] = acc10[v] + bias0;
        out[node1 * OUT_DIM_C + col1] = acc11[v] + bias1;
      }
    }
  }
}

// ---------------- launcher ----------------
extern "C" void kernel_launch(void* const* d_in, const int* in_sizes, int n_in,
                              void* d_out, int out_size, void* d_ws, size_t ws_size,
                              hipStream_t stream) {
  (void)in_sizes; (void)n_in; (void)out_size; (void)ws_size;

  const float* nodes = (const float*)d_in[0];
  const float* edges = (const float*)d_in[1];
  const int*   eidx  = (const int*)d_in[2];   // (2, N_EDGES); row 1 = recv
  const int*   batch = (const int*)d_in[3];
  const float* gg    = (const float*)d_in[4];
  const float* W1    = (const float*)d_in[5];
  const float* b1    = (const float*)d_in[6];
  const float* W2    = (const float*)d_in[7];
  const float* b2    = (const float*)d_in[8];
  float*       out   = (float*)d_out;
  float*       agg   = (float*)d_ws;          // N_NODES*EDGE_DIM fp32 = 19.2 MB scratch

  {
    int total = N_NODES_C * EDGE_DIM_C;
    zero_agg_kernel<<<(total + 255) / 256, 256, 0, stream>>>(agg);
  }
  {
    int total = N_EDGES_C * EDGE_DIM_C;   // 76.8M, fits in int
    scatter_edges_kernel<<<(total + 255) / 256, 256, 0, stream>>>(
        edges, eidx + N_EDGES_C, agg);
  }
  {
    int blocks = (N_NODES_C + 63) / 64;
    mlp_wmma_kernel<<<blocks, 256, 0, stream>>>(
        nodes, agg, batch, gg, W1, b1, W2, b2, out);
  }
}